// RNADE2_85469849190694
// MI455X (gfx1250) — compile-verified
//
#include <hip/hip_runtime.h>
#include <math.h>

// Problem constants
#define BB   2048
#define DD   1024
#define HH   64
#define KK   3
#define SKIPD 4
#define DOUT 1020          // D - SKIP
#define NC   16            // D chunks
#define CD   64            // dims per chunk

typedef __attribute__((ext_vector_type(2))) float v2f;
typedef __attribute__((ext_vector_type(8))) float v8f;

// -------- workspace layout (float offsets); total ~20.3 MB --------
#define WS_XT  0                         // xT[D][B]           : D*B
#define WS_WT  (WS_XT + DD*BB)           // Wt[D][H]           : D*H
#define WS_VP  (WS_WT + DD*HH)           // Vpack[D][32][32]   : D*1024  (B-operand swizzled)
#define WS_BP  (WS_VP + DD*1024)         // bpack[D][16]       : D*16
#define WS_P   (WS_BP + DD*16)           // P[NC][B][H]        : NC*B*H

// ---------------- packing kernels ----------------
__global__ void k_transpose_x(const float* __restrict__ x, float* __restrict__ xT) {
  int idx = blockIdx.x * 256 + threadIdx.x;      // idx = d*B + b
  int b = idx & (BB - 1);
  int d = idx >> 11;
  xT[idx] = x[b * DD + d];
}

__global__ void k_pack_w(const float* __restrict__ W, float* __restrict__ Wt) {
  int idx = blockIdx.x * 256 + threadIdx.x;      // idx = d*H + h
  int h = idx & (HH - 1);
  int d = idx >> 6;
  Wt[idx] = W[h * DD + d];
}

// B-operand order: vb[j] (j = 2g+e) at lane l must hold V[d][h][n],
// h = 4g + e + 2*(l>>4), n = l&15 ; n: 0-2 mean, 3-5 std, 6-8 alpha, 9-15 zero.
__global__ void k_pack_v(const float* __restrict__ Vmean, const float* __restrict__ Valpha,
                         const float* __restrict__ Vstd, float* __restrict__ Vp) {
  int idx = blockIdx.x * 256 + threadIdx.x;      // idx = d*1024 + l*32 + j
  int d = idx >> 10;
  int r = idx & 1023;
  int l = r >> 5;
  int j = r & 31;
  int g = j >> 1, e = j & 1;
  int h = 4 * g + e + 2 * (l >> 4);
  int n = l & 15;
  int base = (d * HH + h) * KK;
  float v = 0.f;
  if (n < 3)      v = Vmean[base + n];
  else if (n < 6) v = Vstd [base + n - 3];
  else if (n < 9) v = Valpha[base + n - 6];
  Vp[idx] = v;
}

__global__ void k_pack_b(const float* __restrict__ bmean, const float* __restrict__ balpha,
                         const float* __restrict__ bstd, float* __restrict__ Bp) {
  int idx = blockIdx.x * 256 + threadIdx.x;      // idx = d*16 + n
  int d = idx >> 4;
  int n = idx & 15;
  float v = 0.f;
  if (n < 3)      v = bmean [d * KK + n];
  else if (n < 6) v = bstd  [d * KK + n - 3];
  else if (n < 9) v = balpha[d * KK + n - 6];
  Bp[idx] = v;
}

// ---------------- chunk sums via WMMA: P[ci][b][h] = sum_{d in chunk} x[b,d]*W[h,d] ----------------
__global__ void __launch_bounds__(32) k_chunksum(const float* __restrict__ x,
                                                 const float* __restrict__ Wt,
                                                 float* __restrict__ P) {
  int ci = blockIdx.x >> 7;          // 16 chunks
  int bt = blockIdx.x & 127;         // 128 row tiles
  int b0 = bt * 16, d0 = ci * CD;
  int l = threadIdx.x;
  int m = l & 15, half = l >> 4, h0 = half * 2;

  const v8f zero = {0.f,0.f,0.f,0.f,0.f,0.f,0.f,0.f};
  v8f acc[4] = {zero, zero, zero, zero};

  const float* xrow = x + (size_t)(b0 + m) * DD + d0;
#pragma unroll
  for (int g = 0; g < 16; ++g) {
    float2 ax = *(const float2*)(xrow + 4 * g + h0);
    v2f A; A.x = ax.x; A.y = ax.y;
    const float* w0p = Wt + (size_t)(d0 + 4 * g + h0) * HH + m;
#pragma unroll
    for (int t = 0; t < 4; ++t) {
      v2f Bv; Bv.x = w0p[t * 16]; Bv.y = w0p[HH + t * 16];
      acc[t] = __builtin_amdgcn_wmma_f32_16x16x4_f32(false, A, false, Bv,
                                                     (short)0, acc[t], false, false);
    }
  }
#pragma unroll
  for (int t = 0; t < 4; ++t)
#pragma unroll
    for (int r = 0; r < 8; ++r)
      P[((size_t)ci * BB + b0 + r + 8 * half) * HH + t * 16 + m] = acc[t][r];
}

// ---------------- exclusive scan over chunks: P[ci] = c + sum_{cj<ci} S[cj] ----------------
__global__ void k_scan(const float* __restrict__ c, float* __restrict__ P) {
  int idx = blockIdx.x * 256 + threadIdx.x;      // idx = b*H + h
  int h = idx & (HH - 1);
  float run = c[h];
#pragma unroll
  for (int ci = 0; ci < NC; ++ci) {
    size_t a = (size_t)ci * BB * HH + idx;
    float t = P[a];
    P[a] = run;
    run += t;
  }
}

// ---------------- fused scan + per-dim WMMA contraction + post ----------------
__global__ void __launch_bounds__(256) k_main(const float* __restrict__ xT,
                                              const float* __restrict__ Wt,
                                              const float* __restrict__ Vp,
                                              const float* __restrict__ Bp,
                                              const float* __restrict__ P,
                                              float* __restrict__ out) {
  __shared__ float LW[CD * HH];   // 16 KB: W chunk, [dl][h]
  __shared__ float LB[CD * 16];   // 4 KB : bias chunk, [dl][n]

  int ci = blockIdx.y;
  int d0 = ci * CD;
  int wave = threadIdx.x >> 5;
  int lane = threadIdx.x & 31;
  int b0 = blockIdx.x * 128 + wave * 16;
  int m = lane & 15, half = lane >> 4, h0 = half * 2;
  int n = m;

  for (int i = threadIdx.x; i < CD * HH; i += 256) LW[i] = Wt[(size_t)d0 * HH + i];
  for (int i = threadIdx.x; i < CD * 16; i += 256) LB[i] = Bp[(size_t)d0 * 16 + i];
  __syncthreads();

  // running state ai in WMMA A layout: a[2g+e] = ai[row m][4g+e+2*half]
  float a[32];
  const float* prow = P + ((size_t)ci * BB + b0 + m) * HH;
#pragma unroll
  for (int g = 0; g < 16; ++g) {
    float2 t = *(const float2*)(prow + 4 * g + h0);
    a[2 * g] = t.x; a[2 * g + 1] = t.y;
  }

  // softmax partner lanes (columns 6,7,8 within each 16-lane half)
  int p1 = lane, p2 = lane;
  if (n >= 6 && n <= 8) {
    int t = n - 6;
    p1 = half * 16 + 6 + ((t + 1) % 3);
    p2 = half * 16 + 6 + ((t + 2) % 3);
  }

  float* outMean  = out;
  float* outStd   = out + (size_t)BB * DOUT * KK;
  float* outAlpha = out + 2 * (size_t)BB * DOUT * KK;

  for (int dl = 0; dl < CD; ++dl) {
    int d = d0 + dl;
    if (d >= SKIPD) {
      // B-operand: 32 swizzled floats per lane
      const float4* vpp = (const float4*)(Vp + (size_t)d * 1024 + lane * 32);
      float4 vv[8];
#pragma unroll
      for (int i = 0; i < 8; ++i) vv[i] = vpp[i];
      const float* vb = (const float*)vv;
      if (d + 1 < DD) __builtin_prefetch(Vp + (size_t)(d + 1) * 1024 + lane * 32, 0, 3);

      v8f acc = {0.f,0.f,0.f,0.f,0.f,0.f,0.f,0.f};
#pragma unroll
      for (int g = 0; g < 16; ++g) {
        v2f A;  A.x = fmaxf(a[2 * g], 0.f);  A.y = fmaxf(a[2 * g + 1], 0.f);
        v2f Bv; Bv.x = vb[2 * g];            Bv.y = vb[2 * g + 1];
        acc = __builtin_amdgcn_wmma_f32_16x16x4_f32(false, A, false, Bv,
                                                    (short)0, acc, false, false);
      }

      float bb = LB[dl * 16 + n];
      int dd = d - SKIPD;
#pragma unroll
      for (int r = 0; r < 8; ++r) {
        float v  = acc[r] + bb;
        float v1 = __shfl(v, p1, 32);
        float v2 = __shfl(v, p2, 32);
        float o;
        if (n < 3) {
          o = v;                                        // mean
        } else if (n < 6) {
          o = 2.0f / (1.0f + __expf(-v)) + 0.6f;        // sigmoid*2 + 0.6
        } else {
          float mx = fmaxf(v, fmaxf(v1, v2));           // softmax over k
          float e0 = __expf(v - mx);
          float s  = e0 + __expf(v1 - mx) + __expf(v2 - mx);
          o = e0 / s;
        }
        if (n < 9) {
          int b = b0 + r + 8 * half;
          float* dst = (n < 3) ? outMean : (n < 6) ? outStd : outAlpha;
          int k = (n < 3) ? n : (n < 6) ? (n - 3) : (n - 6);
          dst[((size_t)b * DOUT + dd) * KK + k] = o;
        }
      }
    }
    // rank-1 update: ai += x[b,d] * W[:,d]   (W broadcast matches A layout via b64 LDS read)
    float xv = xT[(size_t)d * BB + b0 + m];
#pragma unroll
    for (int g = 0; g < 16; ++g) {
      float2 w = *(const float2*)(&LW[dl * HH + 4 * g + h0]);
      a[2 * g]     = fmaf(xv, w.x, a[2 * g]);
      a[2 * g + 1] = fmaf(xv, w.y, a[2 * g + 1]);
    }
  }
}

// ---------------- host launcher ----------------
extern "C" void kernel_launch(void* const* d_in, const int* in_sizes, int n_in,
                              void* d_out, int out_size, void* d_ws, size_t ws_size,
                              hipStream_t stream) {
  (void)in_sizes; (void)n_in; (void)out_size; (void)ws_size;
  const float* x      = (const float*)d_in[0];
  const float* W      = (const float*)d_in[1];
  const float* c      = (const float*)d_in[2];
  const float* Vmean  = (const float*)d_in[3];
  const float* Valpha = (const float*)d_in[4];
  const float* Vstd   = (const float*)d_in[5];
  const float* bmean  = (const float*)d_in[6];
  const float* balpha = (const float*)d_in[7];
  const float* bstd   = (const float*)d_in[8];
  float* out = (float*)d_out;
  float* ws  = (float*)d_ws;

  float* xT = ws + WS_XT;
  float* Wt = ws + WS_WT;
  float* Vp = ws + WS_VP;
  float* Bp = ws + WS_BP;
  float* P  = ws + WS_P;

  k_transpose_x<<<(DD * BB) / 256, 256, 0, stream>>>(x, xT);
  k_pack_w<<<(DD * HH) / 256, 256, 0, stream>>>(W, Wt);
  k_pack_v<<<(DD * 1024) / 256, 256, 0, stream>>>(Vmean, Valpha, Vstd, Vp);
  k_pack_b<<<(DD * 16) / 256, 256, 0, stream>>>(bmean, balpha, bstd, Bp);
  k_chunksum<<<NC * (BB / 16), 32, 0, stream>>>(x, Wt, P);
  k_scan<<<(BB * HH) / 256, 256, 0, stream>>>(c, P);
  k_main<<<dim3(BB / 128, NC), 256, 0, stream>>>(xT, Wt, Vp, Bp, P, out);
}